// CriticGNN_35502199669066
// MI455X (gfx1250) — compile-verified
//
#include <hip/hip_runtime.h>
#include <hip/hip_bf16.h>

typedef float v2f __attribute__((ext_vector_type(2)));
typedef float v8f __attribute__((ext_vector_type(8)));

#define D 64  // feature dim for all layers

// ---------------- degree / norm ----------------
__global__ void k_zero(float* __restrict__ p, int n) {
    int i = blockIdx.x * blockDim.x + threadIdx.x;
    if (i < n) p[i] = 0.0f;
}

__global__ void k_degree(const int* __restrict__ dst, float* __restrict__ deg, int E) {
    int e = blockIdx.x * blockDim.x + threadIdx.x;
    if (e < E) atomicAdd(&deg[dst[e]], 1.0f);
}

__global__ void k_rsqrt(const float* __restrict__ deg, float* __restrict__ disq, int n) {
    int i = blockIdx.x * blockDim.x + threadIdx.x;
    if (i < n) disq[i] = rsqrtf(deg[i] + 1.0f);  // self-loop adds 1
}

// ---------------- dense H = X @ W via V_WMMA_F32_16X16X4_F32 ----------------
// Block = 128 threads = 4 waves; wave w computes the 16x16 tile at
// rows [blockIdx.x*16, +16), cols [w*16, +16). K = 64 -> 16 WMMA steps of K=4.
// A layout (16x4 f32): lanes 0-15: M=lane, VGPR0=K0,VGPR1=K1; lanes 16-31: K2,K3.
// B layout (4x16 f32): VGPR0: half-wave0 K0 / half-wave1 K2; VGPR1: K1 / K3; N=lane&15.
// C/D layout: VGPR r: lanes 0-15 -> M=r, lanes 16-31 -> M=r+8; N=lane&15.
__global__ __launch_bounds__(128)
void k_gemm_wmma(const float* __restrict__ X, const float* __restrict__ W,
                 float* __restrict__ H) {
    const int lane    = threadIdx.x & 31;
    const int wv      = threadIdx.x >> 5;       // 0..3 -> column tile
    const int m       = lane & 15;
    const int koff    = (lane >> 4) << 1;       // 0 or 2
    const int row     = blockIdx.x * 16 + m;    // N % 16 == 0, no tail
    const int colbase = wv * 16;

    const float* __restrict__ xr = X + row * D;
    v8f c = {};
#pragma unroll
    for (int kk = 0; kk < D; kk += 4) {
        v2f a, b;
        a.x = xr[kk + koff];
        a.y = xr[kk + koff + 1];
        b.x = W[(kk + koff    ) * D + colbase + m];
        b.y = W[(kk + koff + 1) * D + colbase + m];
        c = __builtin_amdgcn_wmma_f32_16x16x4_f32(
            /*neg_a=*/false, a, /*neg_b=*/false, b,
            /*c_mod=*/(short)0, c, /*reuse_a=*/false, /*reuse_b=*/false);
    }

    const int nout  = colbase + m;
    const int mbase = blockIdx.x * 16 + ((lane >> 4) << 3);
#pragma unroll
    for (int r = 0; r < 8; ++r)
        H[(mbase + r) * D + nout] = c[r];
}

// ---------------- agg init: A = H * d_ii^2 + bias (full overwrite) ----------------
__global__ void k_init_agg(const float* __restrict__ H, const float* __restrict__ disq,
                           const float* __restrict__ bias, float* __restrict__ A, int n) {
    int t = blockIdx.x * blockDim.x + threadIdx.x;
    if (t >= n * D) return;
    int i = t >> 6, f = t & (D - 1);
    float d = disq[i];
    A[t] = H[t] * d * d + bias[f];
}

// ---------------- edge scatter: one wave per edge, 2 features per lane ----------------
__global__ void k_edge_agg(const float* __restrict__ H, const float* __restrict__ disq,
                           const int* __restrict__ src, const int* __restrict__ dst,
                           float* __restrict__ A, int E) {
    int t = blockIdx.x * blockDim.x + threadIdx.x;
    int e = t >> 5;
    if (e >= E) return;
    int f = (t & 31) * 2;
    int s = src[e], d = dst[e];
    float norm = disq[s] * disq[d];
    float2 hv = *(const float2*)(H + (size_t)s * D + f);
    atomicAdd(&A[(size_t)d * D + f    ], hv.x * norm);
    atomicAdd(&A[(size_t)d * D + f + 1], hv.y * norm);
}

// ---------------- ReLU ----------------
__global__ void k_relu(float* __restrict__ A, int n64) {
    int t = blockIdx.x * blockDim.x + threadIdx.x;
    if (t < n64) A[t] = fmaxf(A[t], 0.0f);
}

// ---------------- output head: out = A @ Wout + bout ----------------
__global__ void k_out(const float* __restrict__ A, const float* __restrict__ Wout,
                      const float* __restrict__ bout, float* __restrict__ out, int n) {
    int i = blockIdx.x * blockDim.x + threadIdx.x;
    if (i >= n) return;
    const float4* a4 = (const float4*)(A + (size_t)i * D);
    const float4* w4 = (const float4*)Wout;
    float acc = 0.0f;
#pragma unroll
    for (int k = 0; k < D / 4; ++k) {
        float4 a = a4[k], w = w4[k];
        acc += a.x * w.x + a.y * w.y + a.z * w.z + a.w * w.w;
    }
    out[i] = acc + bout[0];
}

extern "C" void kernel_launch(void* const* d_in, const int* in_sizes, int n_in,
                              void* d_out, int out_size, void* d_ws, size_t ws_size,
                              hipStream_t stream) {
    const float* x    = (const float*)d_in[0];
    const int*   ei   = (const int*)d_in[1];
    const float* W0   = (const float*)d_in[2];
    const float* b0   = (const float*)d_in[3];
    const float* W1   = (const float*)d_in[4];
    const float* b1   = (const float*)d_in[5];
    const float* W2   = (const float*)d_in[6];
    const float* b2   = (const float*)d_in[7];
    const float* Wout = (const float*)d_in[8];
    const float* bout = (const float*)d_in[9];
    float* out = (float*)d_out;

    const int N = in_sizes[0] / D;       // 50000
    const int E = in_sizes[1] / 2;       // 800000
    const int* src = ei;
    const int* dst = ei + E;

    // workspace layout (256B aligned slices)
    auto align = [](size_t v) { return (v + 255) & ~size_t(255); };
    char* ws = (char*)d_ws;
    size_t off = 0;
    float* deg  = (float*)(ws + off); off = align(off + (size_t)N * 4);
    float* disq = (float*)(ws + off); off = align(off + (size_t)N * 4);
    float* H    = (float*)(ws + off); off = align(off + (size_t)N * D * 4);
    float* A0   = (float*)(ws + off); off = align(off + (size_t)N * D * 4);
    float* A1   = (float*)(ws + off); off = align(off + (size_t)N * D * 4);
    (void)ws_size; (void)n_in; (void)out_size;

    const int T = 256;
    const int gN   = (N + T - 1) / T;
    const int gE   = (E + T - 1) / T;
    const int gNF  = (N * D + T - 1) / T;
    const int gEW  = ((E * 32) + T - 1) / T;   // one wave (32 lanes) per edge
    const int gRT  = N / 16;                   // row tiles for WMMA GEMM

    // degrees + normalization
    k_zero  <<<gN, T, 0, stream>>>(deg, N);
    k_degree<<<gE, T, 0, stream>>>(dst, deg, E);
    k_rsqrt <<<gN, T, 0, stream>>>(deg, disq, N);

    // layer 0: x -> A0
    k_gemm_wmma<<<gRT, 128, 0, stream>>>(x, W0, H);
    k_init_agg <<<gNF, T, 0, stream>>>(H, disq, b0, A0, N);
    k_edge_agg <<<gEW, T, 0, stream>>>(H, disq, src, dst, A0, E);
    k_relu     <<<gNF, T, 0, stream>>>(A0, N * D);

    // layer 1: A0 -> A1
    k_gemm_wmma<<<gRT, 128, 0, stream>>>(A0, W1, H);
    k_init_agg <<<gNF, T, 0, stream>>>(H, disq, b1, A1, N);
    k_edge_agg <<<gEW, T, 0, stream>>>(H, disq, src, dst, A1, E);
    k_relu     <<<gNF, T, 0, stream>>>(A1, N * D);

    // layer 2: A1 -> A0
    k_gemm_wmma<<<gRT, 128, 0, stream>>>(A1, W2, H);
    k_init_agg <<<gNF, T, 0, stream>>>(H, disq, b2, A0, N);
    k_edge_agg <<<gEW, T, 0, stream>>>(H, disq, src, dst, A0, E);
    k_relu     <<<gNF, T, 0, stream>>>(A0, N * D);

    // output head
    k_out<<<gN, T, 0, stream>>>(A0, Wout, bout, out, N);
}